// MiniBatchDiscrimination_41326175322776
// MI455X (gfx1250) — compile-verified
//
#include <hip/hip_runtime.h>
#include <hip/hip_bf16.h>

// MiniBatchDiscrimination for MI455X (gfx1250, wave32).
// N=256, A=1024, B=64, C=16.
//
// Stage 1: M = x @ T.reshape(1024,1024) via V_WMMA_F32_16X16X4_F32
//          (full-precision fp32 matrix pipe; matches fp32 reference).
// Stage 2: per feature-block b: D[k,j] = sum_c |M[k,b,c]-M[j,b,c]|,
//          S[k] = sum_j exp(-D[k,j]), out[k,b] = S[k] - adj
//          where adj = 1 (k==0) else exp(-D[k,255]).

typedef __attribute__((ext_vector_type(2))) float v2f;
typedef __attribute__((ext_vector_type(8))) float v8f;

#define NN 256
#define AA 1024
#define BB 64
#define CC 16
#define BC (BB * CC)   // 1024 columns of M

// ---------------------------------------------------------------------------
// Stage 1: fp32 GEMM, one 16x16 output tile per wave32.
//   Tiles: 16 row-tiles x 64 col-tiles = 1024 waves = 128 blocks x 8 waves.
//   Per K-step of 4: A is 16x4 (lane m = row, half-wave selects K pair),
//   B is 4x16 (lane m = col, half-wave selects K pair), per ISA 7.12.2.
// ---------------------------------------------------------------------------
__global__ void __launch_bounds__(256)
gemm_wmma_f32(const float* __restrict__ x, const float* __restrict__ T,
              float* __restrict__ M) {
  const int lane = threadIdx.x & 31;
  const int wave = threadIdx.x >> 5;
  const int wid  = blockIdx.x * 8 + wave;      // 0..1023

  const int tileRow = wid >> 6;                // 0..15  (rows of M / 16)
  const int tileCol = wid & 63;                // 0..63  (cols of M / 16)
  const int row0 = tileRow * 16;
  const int col0 = tileCol * 16;

  const int m  = lane & 15;                    // A: row-in-tile / B: col-in-tile
  const int kh = (lane >> 4) << 1;             // 0 for lanes 0-15, 2 for 16-31

  const float* __restrict__ xrow = x + (row0 + m) * AA;   // A row
  const float* __restrict__ tcol = T + (col0 + m);        // B column (stride BC)

  v8f acc = {};
#pragma unroll 4
  for (int k0 = 0; k0 < AA; k0 += 4) {
    // A 16x4: VGPR0 = K=kh, VGPR1 = K=kh+1 (8B aligned -> b64 load)
    const v2f a = *(const v2f*)(xrow + k0 + kh);
    // B 4x16: lane's column, K = kh, kh+1 (stride-BC rows; b32 loads that
    // coalesce into 64B segments across each half-wave)
    v2f bm;
    bm.x = tcol[(k0 + kh + 0) * BC];
    bm.y = tcol[(k0 + kh + 1) * BC];
    acc = __builtin_amdgcn_wmma_f32_16x16x4_f32(
        /*neg_a=*/false, a, /*neg_b=*/false, bm,
        /*c_mod=*/(short)0, acc, /*reuse_a=*/false, /*reuse_b=*/false);
  }

  // C/D layout: VGPR r -> row = r + 8*(lane/16), col = lane%16
  const int crow = row0 + ((lane >> 4) << 3);
  const int ccol = col0 + (lane & 15);
#pragma unroll
  for (int r = 0; r < 8; ++r) {
    M[(crow + r) * BC + ccol] = acc[r];
  }
}

// ---------------------------------------------------------------------------
// Stage 2: one block per feature-block b (64 blocks x 256 threads).
//   LDS holds M[:, b, :] (256 x 16 fp32 = 16KB of the 320KB/WGP budget).
//   Thread k keeps its own 16 values in registers, sweeps j with broadcast
//   LDS reads (uniform address across the wave -> no bank conflicts).
// ---------------------------------------------------------------------------
__global__ void __launch_bounds__(256)
pairwise_l1_exp(const float* __restrict__ M, float* __restrict__ out) {
  const int b = blockIdx.x;     // 0..63
  const int k = threadIdx.x;    // 0..255

  __shared__ float lds[NN * CC];   // 16 KB

  // Own row: 64B-aligned -> 4 x global_load_b128
  const float4* __restrict__ src = (const float4*)(M + (k * BB + b) * CC);
  const float4 m0 = src[0], m1 = src[1], m2 = src[2], m3 = src[3];

  float4* dst = (float4*)(lds + k * CC);
  dst[0] = m0; dst[1] = m1; dst[2] = m2; dst[3] = m3;
  __syncthreads();

  float S = 0.0f;
  float last = 0.0f;

#pragma unroll 4
  for (int j = 0; j < NN; ++j) {
    const float4* __restrict__ row = (const float4*)(lds + j * CC);
    const float4 a0 = row[0], a1 = row[1], a2 = row[2], a3 = row[3];
    float d;
    d  = fabsf(m0.x - a0.x) + fabsf(m0.y - a0.y)
       + fabsf(m0.z - a0.z) + fabsf(m0.w - a0.w);
    d += fabsf(m1.x - a1.x) + fabsf(m1.y - a1.y)
       + fabsf(m1.z - a1.z) + fabsf(m1.w - a1.w);
    d += fabsf(m2.x - a2.x) + fabsf(m2.y - a2.y)
       + fabsf(m2.z - a2.z) + fabsf(m2.w - a2.w);
    d += fabsf(m3.x - a3.x) + fabsf(m3.y - a3.y)
       + fabsf(m3.z - a3.z) + fabsf(m3.w - a3.w);
    const float e = __expf(-d);
    S += e;
    if (j == NN - 1) last = e;   // cmat[k, N-1, b]
  }

  // k==0: drop self (exp(0)=1); else drop last column.
  const float adj = (k == 0) ? 1.0f : last;
  out[k * BB + b] = S - adj;
}

// ---------------------------------------------------------------------------
extern "C" void kernel_launch(void* const* d_in, const int* in_sizes, int n_in,
                              void* d_out, int out_size, void* d_ws, size_t ws_size,
                              hipStream_t stream) {
  const float* x = (const float*)d_in[0];   // (256, 1024) fp32
  const float* T = (const float*)d_in[1];   // (1024, 64, 16) fp32
  float* out = (float*)d_out;               // (256, 64) fp32
  float* M   = (float*)d_ws;                // (256, 1024) fp32 scratch (1 MB)

  // 1024 tile-waves / 8 waves per block = 128 blocks
  gemm_wmma_f32<<<128, 256, 0, stream>>>(x, T, M);
  // one block per feature-block b
  pairwise_l1_exp<<<BB, 256, 0, stream>>>(M, out);
}